// EpisodicSlotWriter_12283606467871
// MI455X (gfx1250) — compile-verified
//
#include <hip/hip_runtime.h>
#include <hip/hip_bf16.h>
#include <math.h>
#include <stdint.h>

#define B_DIM 64
#define K_DIM 2048
#define D_DIM 512
#define EPSV 1e-6f
#define DECAY 0.995f
#define MINSTR 0.001f
#define ALPHA 0.5f

typedef float v2f __attribute__((ext_vector_type(2)));
typedef float v4f __attribute__((ext_vector_type(4)));
typedef float v8f __attribute__((ext_vector_type(8)));

// ---------------- Kernel 0: per-batch 1/(||write_key||+eps) ----------------
__global__ void __launch_bounds__(256) k_wk_norm(const float* __restrict__ wk,
                                                 float* __restrict__ inv_nwk) {
    __shared__ float red[256];
    int b = blockIdx.x, t = threadIdx.x;
    const float* row = wk + (size_t)b * D_DIM;
    float acc = 0.f;
    for (int i = t; i < D_DIM; i += 256) { float v = row[i]; acc += v * v; }
    red[t] = acc; __syncthreads();
    for (int s = 128; s > 0; s >>= 1) {
        if (t < s) red[t] += red[t + s];
        __syncthreads();
    }
    if (t == 0) inv_nwk[b] = 1.0f / (sqrtf(red[0]) + EPSV);
}

// ---- Kernel 1: per 16-row tile: normalize keys rows + sim via WMMA f32 ----
// Tile + write_key staged via CDNA5 async global->LDS DMA (no VGPR round-trip).
__global__ void __launch_bounds__(32) k_sim_keys(const float* __restrict__ epi_keys,
                                                 const float* __restrict__ write_key,
                                                 const float* __restrict__ inv_nwk_arr,
                                                 float* __restrict__ keys_out,
                                                 float* __restrict__ sim_out) {
    __shared__ float tile[16 * D_DIM];  // 32 KB
    __shared__ float wkl[D_DIM];        // 2 KB
    __shared__ float invn[16];
    __shared__ float simr[16];

    const int lane = threadIdx.x;              // wave32 block
    const int tileId = blockIdx.x;
    const int b  = tileId >> 7;                // K/16 = 128 tiles per batch
    const int k0 = (tileId & 127) * 16;

    const float* src   = epi_keys  + ((size_t)(b * K_DIM + k0)) * D_DIM;
    const float* wkrow = write_key + (size_t)b * D_DIM;

    // ---- async direct global->LDS (GLOBAL_LOAD_ASYNC_TO_LDS_B128) ----
    {
        uint32_t lds_wk   = (uint32_t)(uintptr_t)wkl;    // wave-relative LDS offset
        uint32_t lds_tile = (uint32_t)(uintptr_t)tile;
        for (int i = 0; i < 4; ++i) {                    // write_key: 128 float4
            int f = i * 32 + lane;
            const float4* g = ((const float4*)wkrow) + f;
            uint32_t l = lds_wk + (uint32_t)f * 16u;
            asm volatile("global_load_async_to_lds_b128 %0, %1, off"
                         :: "v"(l), "v"(g) : "memory");
        }
        for (int i = 0; i < 64; ++i) {                   // tile: 2048 float4
            int f = i * 32 + lane;
            const float4* g = ((const float4*)src) + f;
            uint32_t l = lds_tile + (uint32_t)f * 16u;
            asm volatile("global_load_async_to_lds_b128 %0, %1, off"
                         :: "v"(l), "v"(g) : "memory");
        }
        asm volatile("s_wait_asynccnt 0x0" ::: "memory");
    }
    __syncthreads();

    // per-row 1/(||row||+eps): two lanes per row, half-row each
    {
        int row = lane >> 1, half = lane & 1;
        const float* rp = tile + row * D_DIM + half * 256;
        float acc = 0.f;
        for (int j = 0; j < 256; ++j) { float v = rp[j]; acc += v * v; }
        acc += __shfl_xor(acc, 1, 32);
        if (half == 0) invn[row] = 1.0f / (sqrtf(acc) + EPSV);
    }
    __syncthreads();

#if __has_builtin(__builtin_amdgcn_wmma_f32_16x16x4_f32)
    // sim raw dots via V_WMMA_F32_16X16X4_F32, accumulating over D in K=4 steps.
    // A layout (16x4 f32): lanes 0-15 row M, {VGPR0,VGPR1}={K0,K1}; lanes 16-31: {K2,K3}.
    // B (4x16) carries write_key replicated across all 16 columns.
    {
        const int rowA = lane & 15;
        const int koff = (lane < 16) ? 0 : 2;
        const float* arow = tile + rowA * D_DIM;
        v8f acc = {0.f, 0.f, 0.f, 0.f, 0.f, 0.f, 0.f, 0.f};
        for (int d0 = 0; d0 < D_DIM; d0 += 4) {
            v2f a, bb;
            a.x  = arow[d0 + koff];
            a.y  = arow[d0 + koff + 1];
            bb.x = wkl[d0 + koff];
            bb.y = wkl[d0 + koff + 1];
            acc = __builtin_amdgcn_wmma_f32_16x16x4_f32(false, a, false, bb,
                                                        (short)0, acc, false, false);
        }
        // C/D layout: VGPR v, lanes 0-15 -> M=v ; lanes 16-31 -> M=v+8. Column 0 suffices.
        if (lane == 0)  for (int v = 0; v < 8; ++v) simr[v]     = acc[v];
        if (lane == 16) for (int v = 0; v < 8; ++v) simr[8 + v] = acc[v];
    }
#else
    // VALU fallback dot (two lanes per row)
    {
        int row = lane >> 1, half = lane & 1;
        const float* rp = tile + row * D_DIM + half * 256;
        const float* wp = wkl + half * 256;
        float acc = 0.f;
        for (int j = 0; j < 256; ++j) acc += rp[j] * wp[j];
        acc += __shfl_xor(acc, 1, 32);
        if (half == 0) simr[row] = acc;
    }
#endif
    __syncthreads();

    const float inv_nwk = inv_nwk_arr[b];
    if (lane < 16)
        sim_out[b * K_DIM + k0 + lane] = simr[lane] * invn[lane] * inv_nwk;

    // write normalized keys rows, non-temporal (winner row fixed up later)
    v4f* dst = (v4f*)(keys_out + ((size_t)(b * K_DIM + k0)) * D_DIM);
    for (int i = 0; i < 64; ++i) {
        int f = i * 32 + lane;
        int row = f >> 7;   // 128 float4 per row
        float s = invn[row];
        v4f v = ((const v4f*)tile)[f];
        v = v * s;
        __builtin_nontemporal_store(v, dst + f);
    }
}

// --------------- Kernel 2: per-batch argmax / max of sim -------------------
__global__ void __launch_bounds__(256) k_argmax(const float* __restrict__ sim,
                                                int* __restrict__ slot_ws,
                                                float* __restrict__ slot_out,
                                                float* __restrict__ best_out) {
    __shared__ float vred[256];
    __shared__ int   ired[256];
    int b = blockIdx.x, t = threadIdx.x;
    const float* s = sim + (size_t)b * K_DIM;
    float bv = -INFINITY; int bi = 0x7fffffff;
    for (int k = t; k < K_DIM; k += 256) {
        float v = s[k];
        if (v > bv || (v == bv && k < bi)) { bv = v; bi = k; }
    }
    vred[t] = bv; ired[t] = bi; __syncthreads();
    for (int st = 128; st > 0; st >>= 1) {
        if (t < st) {
            float ov = vred[t + st]; int oi = ired[t + st];
            if (ov > vred[t] || (ov == vred[t] && oi < ired[t])) { vred[t] = ov; ired[t] = oi; }
        }
        __syncthreads();
    }
    if (t == 0) {
        slot_ws[b] = ired[0];
        slot_out[b] = (float)ired[0];
        best_out[b] = vred[0];
    }
}

// ------- Kernel 3: bulk copy epi_vals -> vals_new (non-temporal) -----------
__global__ void __launch_bounds__(256) k_copy_vals(const v4f* __restrict__ src,
                                                   v4f* __restrict__ dst) {
    size_t i = (size_t)blockIdx.x * 256 + threadIdx.x;
    v4f v = __builtin_nontemporal_load(src + i);
    __builtin_nontemporal_store(v, dst + i);
}

// ------------- Kernel 4: blend + renormalize the winning rows --------------
__global__ void __launch_bounds__(128) k_fixup(const float* __restrict__ epi_keys,
                                               const float* __restrict__ epi_vals,
                                               const float* __restrict__ write_key,
                                               const float* __restrict__ write_val,
                                               const float* __restrict__ write_strength,
                                               const int* __restrict__ slot_ws,
                                               float* __restrict__ keys_out,
                                               float* __restrict__ vals_out) {
    __shared__ float red[128];
    int b = blockIdx.x, t = threadIdx.x;
    int slot = slot_ws[b];
    float ws = fminf(fmaxf(write_strength[b], 0.f), 1.f);
    float rate = ws * ALPHA, omr = 1.f - rate;
    size_t off = ((size_t)(b * K_DIM + slot)) * D_DIM;

    float4 ek = ((const float4*)(epi_keys + off))[t];
    float4 wk = ((const float4*)(write_key + (size_t)b * D_DIM))[t];
    float4 kb;
    kb.x = omr * ek.x + rate * wk.x;
    kb.y = omr * ek.y + rate * wk.y;
    kb.z = omr * ek.z + rate * wk.z;
    kb.w = omr * ek.w + rate * wk.w;
    red[t] = kb.x * kb.x + kb.y * kb.y + kb.z * kb.z + kb.w * kb.w;
    __syncthreads();
    for (int s = 64; s > 0; s >>= 1) {
        if (t < s) red[t] += red[t + s];
        __syncthreads();
    }
    float invn = 1.0f / (sqrtf(red[0]) + EPSV);
    kb.x *= invn; kb.y *= invn; kb.z *= invn; kb.w *= invn;
    ((float4*)(keys_out + off))[t] = kb;

    float4 ev = ((const float4*)(epi_vals + off))[t];
    float4 wv = ((const float4*)(write_val + (size_t)b * D_DIM))[t];
    float4 vb;
    vb.x = omr * ev.x + rate * wv.x;
    vb.y = omr * ev.y + rate * wv.y;
    vb.z = omr * ev.z + rate * wv.z;
    vb.w = omr * ev.w + rate * wv.w;
    ((float4*)(vals_out + off))[t] = vb;
}

// ---------------------- Kernel 5: age / strength update --------------------
__global__ void __launch_bounds__(256) k_age_str(const float* __restrict__ age,
                                                 const float* __restrict__ str,
                                                 const float* __restrict__ write_strength,
                                                 const int* __restrict__ slot_ws,
                                                 float* __restrict__ age_out,
                                                 float* __restrict__ str_out) {
    int idx = blockIdx.x * 256 + threadIdx.x;   // B*K total
    int b = idx >> 11;                          // /K_DIM
    int k = idx & (K_DIM - 1);
    float w = (k == slot_ws[b]) ? 1.f : 0.f;
    age_out[idx] = (age[idx] + 1.f) * (1.f - w);
    float ws = fminf(fmaxf(write_strength[b], 0.f), 1.f);
    float s = str[idx] * DECAY;
    s = s + w * ws * (1.f - s);
    str_out[idx] = fminf(fmaxf(s, MINSTR), 1.f);
}

extern "C" void kernel_launch(void* const* d_in, const int* in_sizes, int n_in,
                              void* d_out, int out_size, void* d_ws, size_t ws_size,
                              hipStream_t stream) {
    const float* write_key      = (const float*)d_in[0];
    const float* write_val      = (const float*)d_in[1];
    const float* write_strength = (const float*)d_in[2];
    const float* epi_keys       = (const float*)d_in[3];
    const float* epi_vals       = (const float*)d_in[4];
    const float* epi_age        = (const float*)d_in[5];
    const float* epi_strength   = (const float*)d_in[6];

    float* out = (float*)d_out;
    const size_t nBKD = (size_t)B_DIM * K_DIM * D_DIM;  // 67,108,864
    const size_t nBK  = (size_t)B_DIM * K_DIM;          // 131,072
    float* keys_out = out;
    float* vals_out = out + nBKD;
    float* age_out  = out + 2 * nBKD;          // also used as sim[] scratch
    float* str_out  = age_out + nBK;
    float* slot_out = str_out + nBK;
    float* best_out = slot_out + B_DIM;

    float* inv_nwk = (float*)d_ws;             // B floats
    int*   slot_ws = (int*)((char*)d_ws + B_DIM * sizeof(float));  // B ints

    float* sim_tmp = age_out;                  // B*K floats, overwritten by k_age_str

    k_wk_norm <<<B_DIM, 256, 0, stream>>>(write_key, inv_nwk);
    k_sim_keys<<<B_DIM * (K_DIM / 16), 32, 0, stream>>>(epi_keys, write_key, inv_nwk,
                                                        keys_out, sim_tmp);
    k_argmax  <<<B_DIM, 256, 0, stream>>>(sim_tmp, slot_ws, slot_out, best_out);
    k_copy_vals<<<(unsigned)(nBKD / 4 / 256), 256, 0, stream>>>((const v4f*)epi_vals,
                                                                (v4f*)vals_out);
    k_fixup   <<<B_DIM, 128, 0, stream>>>(epi_keys, epi_vals, write_key, write_val,
                                          write_strength, slot_ws, keys_out, vals_out);
    k_age_str <<<(unsigned)(nBK / 256), 256, 0, stream>>>(epi_age, epi_strength,
                                                          write_strength, slot_ws,
                                                          age_out, str_out);
}